// Discriminator_44779329028358
// MI455X (gfx1250) — compile-verified
//
#include <hip/hip_runtime.h>
#include <cstdint>
#include <cstddef>

typedef __attribute__((ext_vector_type(2))) float v2f;
typedef __attribute__((ext_vector_type(8))) float v8f;

#define B_    64
#define S_    2048
#define D_    256      // D_IN == D_OUT == 256
#define ROWS  128      // 2 span types * 64 batches
#define CHUNKS 8       // S split into 8 chunks of 256

// ---------------------------------------------------------------------------
// Kernel 1: per-batch argmax of start/end scores + gold indices + span counts
// score layout: [2, B, S];  answer_idx layout: [2, B] (int32)
// Rows 0..63  = hard (argmax) spans, rows 64..127 = gold spans.
// ---------------------------------------------------------------------------
__global__ __launch_bounds__(256) void argmax_idx_kernel(
    const float* __restrict__ score, const int* __restrict__ answer_idx,
    int* __restrict__ sArr, int* __restrict__ eArr, float* __restrict__ countF)
{
    __shared__ float sv[256];
    __shared__ int   si[256];
    const int b   = blockIdx.x;     // 0..63
    const int tid = threadIdx.x;

    int res[2];
    for (int which = 0; which < 2; ++which) {
        const float* p = score + ((size_t)which * B_ + b) * S_;
        float best = -3.402823466e38f;
        int   bi   = 0;
        for (int s = tid; s < S_; s += 256) {
            float v = p[s];
            if (v > best) { best = v; bi = s; }   // first-occurrence within stride
        }
        sv[tid] = best; si[tid] = bi;
        __syncthreads();
        for (int off = 128; off > 0; off >>= 1) {
            if (tid < off) {
                float v2 = sv[tid + off]; int i2 = si[tid + off];
                if (v2 > sv[tid] || (v2 == sv[tid] && i2 < si[tid])) {
                    sv[tid] = v2; si[tid] = i2;   // tie -> smaller index (jnp.argmax)
                }
            }
            __syncthreads();
        }
        res[which] = si[0];
        __syncthreads();
    }

    if (tid == 0) {
        const int s0 = res[0], e0 = res[1];
        sArr[b] = s0; eArr[b] = e0;
        const int ch = e0 - s0 + 1;
        countF[b] = (float)(ch > 0 ? ch : 0);
        const int gs = answer_idx[b];          // answer_idx[0][b]
        const int ge = answer_idx[B_ + b];     // answer_idx[1][b]
        sArr[B_ + b] = gs; eArr[B_ + b] = ge;
        const int cg = ge - gs + 1;
        countF[B_ + b] = (float)(cg > 0 ? cg : 0);
    }
}

// ---------------------------------------------------------------------------
// Kernel 2: chunked span-sum of doc_encoding -> partials Xp[CHUNKS][ROWS][D]
// Coalesced: 256 lanes read 256 consecutive floats of one (b,s,:) row.
// ---------------------------------------------------------------------------
__global__ __launch_bounds__(256) void span_partial_kernel(
    const float* __restrict__ doc, const int* __restrict__ sArr,
    const int* __restrict__ eArr, float* __restrict__ Xp)
{
    const int row   = blockIdx.x;   // 0..127
    const int chunk = blockIdx.y;   // 0..7
    const int d     = threadIdx.x;  // 0..255
    const int b     = row & (B_ - 1);

    const int s0 = sArr[row];
    const int e0 = eArr[row];
    const int c0 = chunk * (S_ / CHUNKS);
    const int c1 = c0 + (S_ / CHUNKS) - 1;
    const int lo = (s0 > c0) ? s0 : c0;
    const int hi = (e0 < c1) ? e0 : c1;

    float acc = 0.0f;
    const float* p = doc + ((size_t)b * S_ + lo) * D_ + d;
    for (int s = lo; s <= hi; ++s) { acc += *p; p += D_; }

    Xp[((size_t)chunk * ROWS + row) * D_ + d] = acc;
}

// ---------------------------------------------------------------------------
// Kernel 3: reduce partials -> X[ROWS][D]
// ---------------------------------------------------------------------------
__global__ __launch_bounds__(256) void span_reduce_kernel(
    const float* __restrict__ Xp, float* __restrict__ X)
{
    const int row = blockIdx.x;
    const int d   = threadIdx.x;
    float acc = 0.0f;
#pragma unroll
    for (int c = 0; c < CHUNKS; ++c)
        acc += Xp[((size_t)c * ROWS + row) * D_ + d];
    X[(size_t)row * D_ + d] = acc;
}

// ---------------------------------------------------------------------------
// Kernel 4: out[128,256] = X[128,256] @ W[256,256] + count[m]*bias[n]
// fp32 WMMA: V_WMMA_F32_16X16X4_F32, one 16x16 tile per wave, K stepped by 4.
// A 16x4 f32 layout (ISA 7.12.2): lanes 0-15 hold K=0/1, lanes 16-31 K=2/3.
// C/D layout: VGPR r, lanes 0-15 -> M=r, lanes 16-31 -> M=8+r.
// ---------------------------------------------------------------------------
__global__ __launch_bounds__(256) void wmma_gemm_kernel(
    const float* __restrict__ X, const float* __restrict__ W,
    const float* __restrict__ bias, const float* __restrict__ countF,
    float* __restrict__ out)
{
    const int tid   = threadIdx.x;
    const int lane  = tid & 31;
    const int wave  = tid >> 5;
    const int gw    = blockIdx.x * 8 + wave;  // 0..127 tiles
    const int mTile = gw & 7;                 // 8 tiles of M (128 rows)
    const int nTile = gw >> 3;                // 16 tiles of N (256 cols)
    const int half  = lane >> 4;              // 0: K pair {0,1}, 1: K pair {2,3}
    const int l     = lane & 15;

    const int n = nTile * 16 + l;
    const float bv = bias[n];

    v8f c;
#pragma unroll
    for (int r = 0; r < 8; ++r) {
        const int m = mTile * 16 + half * 8 + r;
        c[r] = countF[m] * bv;                // bias folded into accumulator
    }

    const float2* Arow = reinterpret_cast<const float2*>(X + ((size_t)(mTile * 16 + l)) * D_);

    for (int k0 = 0; k0 < D_; k0 += 4) {
        const int kk = k0 + half * 2;
        const float2 af = Arow[kk >> 1];      // A[m=l][kk], A[m=l][kk+1]
        v2f a; a.x = af.x; a.y = af.y;
        v2f bb;
        bb.x = W[(size_t)kk * D_ + n];        // B[kk][n]
        bb.y = W[(size_t)(kk + 1) * D_ + n];  // B[kk+1][n]
        c = __builtin_amdgcn_wmma_f32_16x16x4_f32(
                /*neg_a=*/false, a, /*neg_b=*/false, bb,
                /*c_mod=*/(short)0, c, /*reuse_a=*/false, /*reuse_b=*/false);
    }

#pragma unroll
    for (int r = 0; r < 8; ++r) {
        const int m = mTile * 16 + half * 8 + r;
        out[(size_t)m * D_ + n] = c[r];       // rows 0-63 hard, 64-127 gold
    }
}

// ---------------------------------------------------------------------------
// Launch
// ---------------------------------------------------------------------------
extern "C" void kernel_launch(void* const* d_in, const int* in_sizes, int n_in,
                              void* d_out, int out_size, void* d_ws, size_t ws_size,
                              hipStream_t stream)
{
    (void)in_sizes; (void)n_in; (void)out_size; (void)ws_size;

    const float* doc        = (const float*)d_in[0];  // [64,2048,256] f32
    const float* score_soft = (const float*)d_in[1];  // [2,64,2048]   f32
    const int*   answer_idx = (const int*)  d_in[2];  // [2,64]        i32
    const float* W_mlp      = (const float*)d_in[3];  // [256,256]     f32
    const float* b_mlp      = (const float*)d_in[4];  // [256]         f32
    float*       out        = (float*)d_out;          // [2,64,256]    f32

    // Workspace layout
    char*  ws     = (char*)d_ws;
    int*   sArr   = (int*)ws;                         // 128 ints
    int*   eArr   = sArr + ROWS;                      // 128 ints
    float* countF = (float*)(eArr + ROWS);            // 128 floats
    float* Xp     = (float*)(ws + 2048);              // [8][128][256] f32 (1 MB)
    float* X      = Xp + (size_t)CHUNKS * ROWS * D_;  // [128][256]    f32 (128 KB)

    argmax_idx_kernel<<<B_, 256, 0, stream>>>(score_soft, answer_idx, sArr, eArr, countF);
    span_partial_kernel<<<dim3(ROWS, CHUNKS), 256, 0, stream>>>(doc, sArr, eArr, Xp);
    span_reduce_kernel<<<ROWS, 256, 0, stream>>>(Xp, X);
    wmma_gemm_kernel<<<ROWS / 8, 256, 0, stream>>>(X, W_mlp, b_mlp, countF, out);
}